// AllPairs_19430432047334
// MI455X (gfx1250) — compile-verified
//
#include <hip/hip_runtime.h>
#include <hip/hip_bf16.h>

// Problem constants (from the reference: B=1024, C=128 balanced classes)
#define BB   1024
#define CC   128
#define SS   8                    // samples per class = B / C
#define NPOS (BB * (SS - 1))      // 7168
#define NNEG (BB - SS)            // 1016
#define NOUT (NPOS * NNEG)        // 7,282,688  (== 7112 * 1024, exact grid)

typedef int v4i __attribute__((ext_vector_type(4)));

// ---------------------------------------------------------------------------
// Phase 1a: scatter each sample index into its class's member list (unordered).
// One block, 1024 threads, LDS atomics (ds_add_rtn_u32) -> O(1) depth instead
// of the previous 1024-iteration serial scan.
// ---------------------------------------------------------------------------
__global__ __launch_bounds__(1024) void scatter_members_kernel(
    const int* __restrict__ labels, int* __restrict__ members) {
    __shared__ int s_count[CC];
    int i = threadIdx.x;                 // sample index, 0..1023
    if (i < CC) s_count[i] = 0;
    __syncthreads();
    int l = labels[i];
    int slot = atomicAdd(&s_count[l], 1);   // LDS atomic, slot in [0,8)
    members[l * SS + slot] = i;
}

// ---------------------------------------------------------------------------
// Phase 1b: sort each class's 8 member indices ascending.
// Batcher odd-even merge network for n=8 (19 comparators), branchless.
// ---------------------------------------------------------------------------
__global__ void sort_members_kernel(int* __restrict__ members) {
    int c = threadIdx.x;
    if (c >= CC) return;
    int v[SS];
    #pragma unroll
    for (int m = 0; m < SS; ++m) v[m] = members[c * SS + m];

    #define CSWAP(i, j) { int lo = min(v[i], v[j]); int hi = max(v[i], v[j]); \
                          v[i] = lo; v[j] = hi; }
    CSWAP(0,1) CSWAP(2,3) CSWAP(4,5) CSWAP(6,7)
    CSWAP(0,2) CSWAP(1,3) CSWAP(4,6) CSWAP(5,7)
    CSWAP(1,2) CSWAP(5,6)
    CSWAP(0,4) CSWAP(1,5) CSWAP(2,6) CSWAP(3,7)
    CSWAP(2,4) CSWAP(3,5)
    CSWAP(1,2) CSWAP(3,4) CSWAP(5,6)
    #undef CSWAP

    #pragma unroll
    for (int m = 0; m < SS; ++m) members[c * SS + m] = v[m];
}

// ---------------------------------------------------------------------------
// Phase 2: stream out (anchor_idx, pos_idx, neg_idx). One thread produces 4
// consecutive flat outputs (one int4 per array). Store-bandwidth bound:
// 87.4 MB total -> ~3.75us at 23.3 TB/s. LUT staged to LDS via CDNA5 async
// global->LDS b128 loads (ASYNCcnt path), outputs via nontemporal b128 stores.
// ---------------------------------------------------------------------------
__global__ __launch_bounds__(256) void gen_triplets_kernel(
    const int* __restrict__ labels,
    const int* __restrict__ members,
    int* __restrict__ out)
{
    __shared__ __align__(16) int s_members[CC * SS];   // 4 KB LUT

    // Async stage: 256 lanes x 16B = 4096B = whole LUT, one b128 per lane.
    {
        unsigned lds_off =
            (unsigned)(size_t)(&s_members[0]) + threadIdx.x * 16u;
        unsigned long long gaddr =
            (unsigned long long)(size_t)members + (unsigned long long)threadIdx.x * 16ull;
        asm volatile("global_load_async_to_lds_b128 %0, %1, off"
                     :: "v"(lds_off), "v"(gaddr)
                     : "memory");
        asm volatile("s_wait_asynccnt 0" ::: "memory");
    }
    __syncthreads();

    int g = blockIdx.x * blockDim.x + threadIdx.x;     // int4-group index
    if (g * 4 >= NOUT) return;                         // exact grid; safety only

    // Decompose: NNEG = 1016 is divisible by 4 -> groups never straddle rows.
    const int GPR = NNEG / 4;                          // 254 groups per pair-row
    int p = g / GPR;                                   // positive-pair index
    int t = (g - p * GPR) * 4;                         // first negative ordinal
    int a = p / (SS - 1);                              // anchor index
    int k = p - a * (SS - 1);                          // positive ordinal 0..6
    int l = labels[a];                                 // anchor class (L0-hot)
    const int base = l * SS;

    int mem[SS];
    #pragma unroll
    for (int m = 0; m < SS; ++m) mem[m] = s_members[base + m];

    // rank of the anchor inside its sorted class list; skip it for positives
    int rank = 0;
    #pragma unroll
    for (int m = 0; m < SS; ++m) rank += (mem[m] < a) ? 1 : 0;
    int pos = mem[k + ((k >= rank) ? 1 : 0)];

    // t-th..(t+3)-th elements of the complement of mem[] in [0, B):
    // walk the 8 sorted skips ascending, bump candidate when skip <= candidate.
    int neg[4];
    #pragma unroll
    for (int j = 0; j < 4; ++j) {
        int x = t + j;
        #pragma unroll
        for (int m = 0; m < SS; ++m) x += (mem[m] <= x) ? 1 : 0;
        neg[j] = x;
    }

    v4i* out_a = reinterpret_cast<v4i*>(out);
    v4i* out_p = out_a + (NOUT / 4);
    v4i* out_n = out_p + (NOUT / 4);

    v4i va = { a, a, a, a };
    v4i vp = { pos, pos, pos, pos };
    v4i vn = { neg[0], neg[1], neg[2], neg[3] };

    // Output is written once and never re-read: nontemporal b128 stores.
    __builtin_nontemporal_store(va, &out_a[g]);
    __builtin_nontemporal_store(vp, &out_p[g]);
    __builtin_nontemporal_store(vn, &out_n[g]);
}

// ---------------------------------------------------------------------------
// Harness entry point.
//   d_in[0] = embeddings (1024*128 f32, unused by the reference's outputs)
//   d_in[1] = labels     (1024 int32)
//   d_out   = [anchor_idx | pos_idx | neg_idx], 3 * 7,282,688 int32
//   d_ws    = 4 KB members LUT scratch
// ---------------------------------------------------------------------------
extern "C" void kernel_launch(void* const* d_in, const int* in_sizes, int n_in,
                              void* d_out, int out_size, void* d_ws, size_t ws_size,
                              hipStream_t stream) {
    (void)in_sizes; (void)n_in; (void)out_size; (void)ws_size;

    const int* labels = (const int*)d_in[1];
    int* members      = (int*)d_ws;           // CC*SS*4 = 4 KB
    int* out          = (int*)d_out;

    scatter_members_kernel<<<1, 1024, 0, stream>>>(labels, members);
    sort_members_kernel<<<1, 128, 0, stream>>>(members);

    // NOUT / (4 outputs/thread * 256 threads/block) = 7112 blocks exactly
    gen_triplets_kernel<<<NOUT / 1024, 256, 0, stream>>>(labels, members, out);
}